// APPNP_26740466385346
// MI455X (gfx1250) — compile-verified
//
#include <hip/hip_runtime.h>
#include <stdint.h>

#define D_FEAT   64
#define ALPHA    0.1f
#define K_LAYERS 10
#define BLOCK    256
#define EPB      256   // edges staged per block in the scatter kernel

// ---------------------------------------------------------------------------
// CDNA5 async global->LDS copy (ASYNCcnt path, gfx1250-specific).
// VGPR0 = workgroup-relative LDS byte address (low 32 bits of the generic
// pointer to a __shared__ object), VGPR pair = 64-bit global address.
// ---------------------------------------------------------------------------
__device__ __forceinline__ void async_g2l_b128(void* lds_ptr, const void* gptr) {
  uint32_t lds_addr = (uint32_t)(uintptr_t)lds_ptr;  // low 32 bits == LDS offset
  asm volatile("global_load_async_to_lds_b128 %0, %1, off"
               :: "v"(lds_addr), "v"(gptr)
               : "memory");
}

__device__ __forceinline__ void wait_async0() {
#if __has_builtin(__builtin_amdgcn_s_wait_asynccnt)
  __builtin_amdgcn_s_wait_asynccnt(0);
#else
  asm volatile("s_wait_asynccnt 0x0" ::: "memory");
#endif
}

__device__ __forceinline__ void atomic_add_f32(float* p, float v) {
#if defined(__HIP_DEVICE_COMPILE__)
  unsafeAtomicAdd(p, v);   // native global_atomic_add_f32 (no CAS loop)
#else
  atomicAdd(p, v);
#endif
}

// ---------------------------------------------------------------------------
// Utility kernels
// ---------------------------------------------------------------------------
__global__ void zero_f4_kernel(float4* __restrict__ p, int n4) {
  int i = blockIdx.x * blockDim.x + threadIdx.x;
  if (i < n4) p[i] = make_float4(0.f, 0.f, 0.f, 0.f);
}

__global__ void zero_f32_kernel(float* __restrict__ p, int n) {
  int i = blockIdx.x * blockDim.x + threadIdx.x;
  if (i < n) p[i] = 0.f;
}

__global__ void copy_f4_kernel(const float4* __restrict__ s, float4* __restrict__ d, int n4) {
  int i = blockIdx.x * blockDim.x + threadIdx.x;
  if (i < n4) d[i] = s[i];
}

// deg_out/deg_in accumulated as float directly (avg degree 16 -> exact in f32)
__global__ void degree_kernel(const int* __restrict__ src, const int* __restrict__ dst,
                              float* __restrict__ deg_out, float* __restrict__ deg_in,
                              int E) {
  int e = blockIdx.x * blockDim.x + threadIdx.x;
  if (e < E) {
    atomic_add_f32(&deg_out[src[e]], 1.0f);
    atomic_add_f32(&deg_in [dst[e]], 1.0f);
  }
}

// In-place: deg -> clip(deg,1)^(-1/2)
__global__ void norm_kernel(float* __restrict__ a, float* __restrict__ b, int n) {
  int i = blockIdx.x * blockDim.x + threadIdx.x;
  if (i < n) {
    float da = fmaxf(a[i], 1.0f);
    float db = fmaxf(b[i], 1.0f);
    a[i] = 1.0f / sqrtf(da);
    b[i] = 1.0f / sqrtf(db);
  }
}

// ---------------------------------------------------------------------------
// Scatter: agg[dst[e]] += h[src[e]] * norm_src[src[e]]
// 16 lanes per edge, float4 per lane (b128 gather + 4x global_atomic_add_f32).
// Edge index streams staged to LDS with gfx1250 async copies. agg must be
// zero on entry (guaranteed by the initial zero pass / fused zeroing in
// update_kernel).
// ---------------------------------------------------------------------------
__global__ __launch_bounds__(BLOCK)
void scatter_kernel(const int* __restrict__ src, const int* __restrict__ dst,
                    const float* __restrict__ h, const float* __restrict__ norm_src,
                    float* __restrict__ agg, int E) {
  __shared__ int ls_src[EPB];
  __shared__ int ls_dst[EPB];

  const int  tid  = threadIdx.x;
  const long base = (long)blockIdx.x * EPB;
  const bool full = (base + EPB) <= (long)E;

  if (full) {
    // 256 ints per stream = 64 b128 async transfers each; waves 0..1 issue
    // src, waves 2..3 issue dst, all overlapped and tracked by ASYNCcnt.
    if (tid < EPB / 4) {
      async_g2l_b128(&ls_src[tid * 4], src + base + tid * 4);
    } else if (tid < EPB / 2) {
      int t = tid - EPB / 4;
      async_g2l_b128(&ls_dst[t * 4], dst + base + t * 4);
    }
    wait_async0();           // waves that issued nothing pass immediately
  } else {                   // guarded tail path
    for (int i = tid; i < EPB; i += BLOCK) {
      long e = base + i;
      ls_src[i] = (e < (long)E) ? src[e] : 0;
      ls_dst[i] = (e < (long)E) ? dst[e] : 0;
    }
  }
  __syncthreads();

  const int c   = tid & 15;   // float4 chunk within the 64-wide feature row
  const int sub = tid >> 4;   // which of the 16 edges handled per pass

#pragma unroll 4
  for (int it = 0; it < EPB / (BLOCK / 16); ++it) {
    const int  eloc = it * (BLOCK / 16) + sub;
    const long eg   = base + eloc;
    if (eg >= (long)E) break;

    const int s = ls_src[eloc];
    const int d = ls_dst[eloc];
    const float ns = norm_src[s];   // same addr across the 16-lane group

    const float4 v = *reinterpret_cast<const float4*>(h + (size_t)s * D_FEAT + c * 4);
    float* ap = agg + (size_t)d * D_FEAT + c * 4;
    atomic_add_f32(ap + 0, v.x * ns);
    atomic_add_f32(ap + 1, v.y * ns);
    atomic_add_f32(ap + 2, v.z * ns);
    atomic_add_f32(ap + 3, v.w * ns);
  }
}

// h = 0.9 * agg * norm_dst + 0.1 * h0  (float4 per thread), and re-zero agg
// in the same pass so the next layer's scatter starts from a clean buffer
// (saves a dedicated 25.6 MB zeroing sweep + launch per layer).
__global__ __launch_bounds__(BLOCK)
void update_kernel(float* __restrict__ agg, const float* __restrict__ h0,
                   const float* __restrict__ norm_dst, float* __restrict__ h,
                   int N) {
  int t = blockIdx.x * blockDim.x + threadIdx.x;
  int i = t >> 4;
  int c = t & 15;
  if (i < N) {
    const float nd = norm_dst[i];
    const size_t off = (size_t)i * D_FEAT + c * 4;
    const float4 a = *reinterpret_cast<const float4*>(agg + off);
    const float4 f = *reinterpret_cast<const float4*>(h0 + off);
    float4 r;
    r.x = (1.0f - ALPHA) * (a.x * nd) + ALPHA * f.x;
    r.y = (1.0f - ALPHA) * (a.y * nd) + ALPHA * f.y;
    r.z = (1.0f - ALPHA) * (a.z * nd) + ALPHA * f.z;
    r.w = (1.0f - ALPHA) * (a.w * nd) + ALPHA * f.w;
    *reinterpret_cast<float4*>(h + off) = r;
    *reinterpret_cast<float4*>(agg + off) = make_float4(0.f, 0.f, 0.f, 0.f);
  }
}

// ---------------------------------------------------------------------------
extern "C" void kernel_launch(void* const* d_in, const int* in_sizes, int n_in,
                              void* d_out, int out_size, void* d_ws, size_t ws_size,
                              hipStream_t stream) {
  const float* features = (const float*)d_in[0];
  const int*   src      = (const int*)d_in[1];
  const int*   dst      = (const int*)d_in[2];

  const int E = in_sizes[1];
  const int N = in_sizes[0] / D_FEAT;

  float* h        = (float*)d_out;                  // h lives in d_out
  float* norm_src = (float*)d_ws;                   // N floats (deg_out -> norm)
  float* norm_dst = norm_src + N;                   // N floats (deg_in  -> norm)
  float* agg      = norm_dst + N;                   // N*64 floats (16B aligned)

  const int n4    = N * (D_FEAT / 4);               // float4 count of a feature matrix
  const int gE    = (E + BLOCK - 1) / BLOCK;
  const int gN    = (N + BLOCK - 1) / BLOCK;
  const int g2N   = (2 * N + BLOCK - 1) / BLOCK;
  const int gF4   = (n4 + BLOCK - 1) / BLOCK;
  const int gScat = (E + EPB - 1) / EPB;

  // degrees -> norms (one-time per call)
  zero_f32_kernel<<<g2N, BLOCK, 0, stream>>>(norm_src, 2 * N);
  degree_kernel  <<<gE,  BLOCK, 0, stream>>>(src, dst, norm_src, norm_dst, E);
  norm_kernel    <<<gN,  BLOCK, 0, stream>>>(norm_src, norm_dst, N);

  // h = features; agg = 0 (ws is poisoned by the harness)
  copy_f4_kernel<<<gF4, BLOCK, 0, stream>>>((const float4*)features, (float4*)h, n4);
  zero_f4_kernel<<<gF4, BLOCK, 0, stream>>>((float4*)agg, n4);

  for (int k = 0; k < K_LAYERS; ++k) {
    scatter_kernel<<<gScat, BLOCK, 0, stream>>>(src, dst, h, norm_src, agg, E);
    update_kernel <<<gF4, BLOCK, 0, stream>>>(agg, features, norm_dst, h, N);
  }
}